// CRF_33981781246358
// MI455X (gfx1250) — compile-verified
//
#include <hip/hip_runtime.h>
#include <math.h>

typedef __attribute__((ext_vector_type(2))) float v2f;
typedef __attribute__((ext_vector_type(8))) float v8f;

#define B_  256
#define S_  512
#define T_  128
#define RS  132   // padded LDS row stride (floats): 132 mod 64 == 4 -> no bank conflicts

// ---------------------------------------------------------------------------
// Kernel 1: gold-path score per batch (numerator). Also zeroes d_out[0].
// ---------------------------------------------------------------------------
__global__ void crf_score_kernel(const float* __restrict__ emissions,
                                 const long long* __restrict__ tags,
                                 const unsigned char* __restrict__ mask,
                                 const float* __restrict__ transitions,
                                 const float* __restrict__ start_transitions,
                                 const float* __restrict__ end_transitions,
                                 float* __restrict__ numer,
                                 float* __restrict__ out)
{
    int b = blockIdx.x * blockDim.x + threadIdx.x;
    if (b == 0) out[0] = 0.0f;
    if (b >= B_) return;

    const long long*      tg = tags + (size_t)b * S_;
    const unsigned char*  mk = mask + (size_t)b * S_;
    const float*          em = emissions + (size_t)b * S_ * T_;

    int t0 = (int)tg[0];
    float score = start_transitions[t0] + em[t0];
    int cnt = mk[0] ? 1 : 0;
    for (int s = 1; s < S_; ++s) {
        int tc = (int)tg[s];
        int tp = (int)tg[s - 1];
        float term = transitions[tc * T_ + tp] + em[s * T_ + tc];
        if (mk[s]) { score += term; ++cnt; }
    }
    int last_tag = (int)tg[cnt - 1];
    numer[b] = score + end_transitions[last_tag];
}

// ---------------------------------------------------------------------------
// Kernel 2: forward-algorithm normalizer via fp32 WMMA in the exp domain.
// One workgroup = 16 batches; 8 waves, each owning a 16-wide tag-column tile.
// B-fragments = exp(transitions) slice, held in 64 VGPRs for all 512 steps.
// ---------------------------------------------------------------------------
__global__ void __launch_bounds__(256, 1)
crf_forward_kernel(const float* __restrict__ emissions,
                   const unsigned char* __restrict__ mask,
                   const float* __restrict__ transitions,
                   const float* __restrict__ start_transitions,
                   const float* __restrict__ end_transitions,
                   const float* __restrict__ numer,
                   float* __restrict__ out)
{
    __shared__ float sScore[16 * RS];   // log-domain forward scores
    __shared__ float sAexp [16 * RS];   // exp(score - max): WMMA A operand
    __shared__ float sMax  [16];        // per-batch running max
    __shared__ float sPart [16 * 16];   // reduction scratch

    const int tid    = threadIdx.x;
    const int lane   = tid & 31;
    const int wave   = tid >> 5;        // 0..7 -> column tile
    const int j0     = wave * 16;
    const int batch0 = blockIdx.x * 16;

    // --- Preload B fragments: B[k][n] = exp(transitions[k][j0+n]), K=128.
    // f32 4x16 B-fragment per chunk c: lane L -> N = L%16, K = 4c + 2*(L/16) (+1)
    const int nB = lane & 15;
    const int kh = (lane >> 4) * 2;
    v2f bfrag[32];
    for (int c = 0; c < 32; ++c) {
        int k = 4 * c + kh;
        bfrag[c].x = __expf(transitions[k * T_ + (j0 + nB)]);
        bfrag[c].y = __expf(transitions[(k + 1) * T_ + (j0 + nB)]);
    }

    // --- Initialize score[b][j] = start_transitions[j] + emissions[b][0][j]
    for (int idx = tid; idx < 16 * T_; idx += 256) {
        int b = idx >> 7;
        int j = idx & 127;
        sScore[b * RS + j] = start_transitions[j]
                           + emissions[(size_t)(batch0 + b) * S_ * T_ + j];
    }
    __syncthreads();

    const int bR  = tid & 15;           // batch row for reductions
    const int seg = tid >> 4;           // 8-wide column segment (0..15)

    for (int s = 1; s < S_; ++s) {
        // (a) per-batch max over 128 tags (two-phase)
        {
            const float* r = &sScore[bR * RS + seg * 8];
            float m = r[0];
            #pragma unroll
            for (int k = 1; k < 8; ++k) m = fmaxf(m, r[k]);
            sPart[bR * 16 + seg] = m;
        }
        __syncthreads();
        if (tid < 16) {
            float m = sPart[tid * 16];
            #pragma unroll
            for (int k = 1; k < 16; ++k) m = fmaxf(m, sPart[tid * 16 + k]);
            sMax[tid] = m;
        }
        __syncthreads();

        // (b) A = exp(score - max)
        {
            float m = sMax[bR];
            const float* src = &sScore[bR * RS + seg * 8];
            float*       dst = &sAexp [bR * RS + seg * 8];
            #pragma unroll
            for (int k = 0; k < 8; ++k) dst[k] = __expf(src[k] - m);
        }
        __syncthreads();

        // (c) C(16x16) = A(16x128) * expT(128x16) via 32 chained f32 WMMAs
        v8f c = {0.f, 0.f, 0.f, 0.f, 0.f, 0.f, 0.f, 0.f};
        {
            const int rowA = lane & 15;
            #pragma unroll
            for (int cc = 0; cc < 32; ++cc) {
                const float* ap = &sAexp[rowA * RS + 4 * cc + kh];
                v2f a; a.x = ap[0]; a.y = ap[1];
                c = __builtin_amdgcn_wmma_f32_16x16x4_f32(
                        false, a, false, bfrag[cc], (short)0, c, false, false);
            }
        }

        // (d) score' = log(C) + max + emission, masked write-back.
        // C layout: VGPR v -> M = v + 8*(lane/16), N = lane%16
        {
            const int n     = lane & 15;
            const int mbase = (lane >> 4) * 8;

            // Prefetch next step's emission tile (gfx1250 global_prefetch_b8):
            // lane L covers batch row mbase + (n&7); collectively all 16 rows
            // of this wave's 64B column segment are pulled a full WMMA chain
            // ahead of use. No counter, no VGPR return.
            if (s + 1 < S_) {
                int prow = mbase + (n & 7);
                const float* pf =
                    &emissions[((size_t)(batch0 + prow) * S_ + (s + 1)) * T_ + j0 + n];
                __builtin_prefetch(pf, 0, 1);
            }

            #pragma unroll
            for (int v = 0; v < 8; ++v) {
                int M = mbase + v;
                size_t gb = (size_t)(batch0 + M);
                float em  = emissions[(gb * S_ + s) * T_ + j0 + n];
                float nxt = __logf(c[v]) + sMax[M] + em;
                float old = sScore[M * RS + j0 + n];
                bool  mk  = mask[gb * S_ + s] != 0;
                sScore[M * RS + j0 + n] = mk ? nxt : old;
            }
        }
        __syncthreads();
    }

    // --- denominator_b = LSE_j(score[b][j] + end_transitions[j])
    {
        const float* r = &sScore[bR * RS + seg * 8];
        float m = r[0] + end_transitions[seg * 8];
        #pragma unroll
        for (int k = 1; k < 8; ++k)
            m = fmaxf(m, r[k] + end_transitions[seg * 8 + k]);
        sPart[bR * 16 + seg] = m;
    }
    __syncthreads();
    if (tid < 16) {
        float m = sPart[tid * 16];
        #pragma unroll
        for (int k = 1; k < 16; ++k) m = fmaxf(m, sPart[tid * 16 + k]);
        sMax[tid] = m;
    }
    __syncthreads();
    {
        float m = sMax[bR];
        const float* r = &sScore[bR * RS + seg * 8];
        float acc = 0.f;
        #pragma unroll
        for (int k = 0; k < 8; ++k)
            acc += __expf(r[k] + end_transitions[seg * 8 + k] - m);
        sPart[bR * 16 + seg] = acc;
    }
    __syncthreads();
    if (tid < 16) {
        float acc = 0.f;
        #pragma unroll
        for (int k = 0; k < 16; ++k) acc += sPart[tid * 16 + k];
        float denom = sMax[tid] + __logf(acc);
        sPart[tid] = denom - numer[batch0 + tid];
    }
    __syncthreads();
    if (tid == 0) {
        float acc = 0.f;
        #pragma unroll
        for (int k = 0; k < 16; ++k) acc += sPart[k];
        atomicAdd(out, acc * (1.0f / (float)B_));
    }
}

// ---------------------------------------------------------------------------
extern "C" void kernel_launch(void* const* d_in, const int* in_sizes, int n_in,
                              void* d_out, int out_size, void* d_ws, size_t ws_size,
                              hipStream_t stream)
{
    const float*         emissions         = (const float*)d_in[0];
    const long long*     tags              = (const long long*)d_in[1];
    const unsigned char* mask              = (const unsigned char*)d_in[2];
    const float*         transitions       = (const float*)d_in[3];
    const float*         start_transitions = (const float*)d_in[4];
    const float*         end_transitions   = (const float*)d_in[5];
    float* out   = (float*)d_out;
    float* numer = (float*)d_ws;            // B floats of scratch

    crf_score_kernel<<<1, 256, 0, stream>>>(emissions, tags, mask, transitions,
                                            start_transitions, end_transitions,
                                            numer, out);
    crf_forward_kernel<<<B_ / 16, 256, 0, stream>>>(emissions, mask, transitions,
                                                    start_transitions, end_transitions,
                                                    numer, out);
}